// GCN_27350351741543
// MI455X (gfx1250) — compile-verified
//
#include <hip/hip_runtime.h>

typedef __attribute__((ext_vector_type(2))) float v2f;
typedef __attribute__((ext_vector_type(8))) float v8f;

#define FDIM 64

// ---------------------------------------------------------------------------
// Degree / normalization
// ---------------------------------------------------------------------------
__global__ void init_ones(float* p, int n) {
  int i = blockIdx.x * blockDim.x + threadIdx.x;
  if (i < n) p[i] = 1.0f;  // self-loop contributes 1 to every node's degree
}

__global__ void deg_count(const int* __restrict__ col, float* deg, int e) {
  int i = blockIdx.x * blockDim.x + threadIdx.x;
  if (i < e) atomicAdd(&deg[col[i]], 1.0f);
}

__global__ void finalize_dinv(float* deg, int n) {
  int i = blockIdx.x * blockDim.x + threadIdx.x;
  if (i < n) {
    float d = deg[i];
    deg[i] = (d > 0.0f) ? rsqrtf(d) : 0.0f;
  }
}

__global__ void zero_f(float* p, size_t n) {
  size_t i = (size_t)blockIdx.x * blockDim.x + threadIdx.x;
  if (i < n) p[i] = 0.0f;
}

// ---------------------------------------------------------------------------
// H = X @ W   (X: [nrows,64] f32 row-major, W: [64,64] f32 row-major)
// One wave computes a 16x64 output slab via 16 k-steps x 4 col-tiles of
// V_WMMA_F32_16X16X4_F32.  W is staged in LDS pre-packed in (k,k+1) pairs:
//   Wp[(k/2)*128 + n*2 + (k&1)] = W[k][n]
// so every B fragment is ONE aligned ds_load_b64 landing directly in the
// even-aligned VGPR pair WMMA wants (no v_dual_mov marshaling).
// 256 threads = 8 waves = 128 rows/block; nrows % 16 == 0; per-wave guard is
// wave-uniform so EXEC is all-1s around the WMMAs.
// ---------------------------------------------------------------------------
__global__ void __launch_bounds__(256) gemm64_wmma(const float* __restrict__ X,
                                                   const float* __restrict__ W,
                                                   float* __restrict__ H,
                                                   int nrows) {
  __shared__ __attribute__((aligned(16))) float Wp[FDIM * FDIM];  // 16 KB packed
  int t = threadIdx.x;

  // Pack W into pair-interleaved LDS layout: 2048 (kpair,n) entries, 8/thread.
  // Global reads stay coalesced over n; LDS writes are aligned float2.
#pragma unroll
  for (int j = 0; j < 8; ++j) {
    int i = t + 256 * j;        // 0..2047
    int kp = i >> 6;            // k/2: 0..31
    int n = i & 63;
    float2 w;
    w.x = W[kp * 128 + n];       // W[2kp  ][n]
    w.y = W[kp * 128 + 64 + n];  // W[2kp+1][n]
    *(float2*)&Wp[kp * 128 + n * 2] = w;
  }
  __syncthreads();

  int wave = t >> 5;
  int lane = t & 31;
  int row0 = blockIdx.x * 128 + wave * 16;  // wave-uniform
  if (row0 >= nrows) return;

  int m = lane & 15;            // A-matrix row owned by this lane
  int khalf = (lane >> 4) * 2;  // K sub-offset per half-wave (A & B layout)
  int n = lane & 15;            // C/D column owned by this lane
  int mD = (lane >> 4) * 8;     // C/D row offset per half-wave

  v8f acc0 = {}, acc1 = {}, acc2 = {}, acc3 = {};
  const float* xrow = X + (size_t)(row0 + m) * FDIM;

#pragma unroll
  for (int kt = 0; kt < 16; ++kt) {
    int k = kt * 4 + khalf;     // even
    v2f a = *(const v2f*)(xrow + k);                      // {X[m][k], X[m][k+1]}
    const v2f* bp = (const v2f*)(Wp + (k >> 1) * 128 + n * 2);
    v2f b0 = bp[0];    // cols n+0  : {W[k][n],    W[k+1][n]}
    v2f b1 = bp[16];   // cols n+16
    v2f b2 = bp[32];   // cols n+32
    v2f b3 = bp[48];   // cols n+48

    acc0 = __builtin_amdgcn_wmma_f32_16x16x4_f32(false, a, false, b0, (short)0, acc0, false, false);
    acc1 = __builtin_amdgcn_wmma_f32_16x16x4_f32(false, a, false, b1, (short)0, acc1, false, false);
    acc2 = __builtin_amdgcn_wmma_f32_16x16x4_f32(false, a, false, b2, (short)0, acc2, false, false);
    acc3 = __builtin_amdgcn_wmma_f32_16x16x4_f32(false, a, false, b3, (short)0, acc3, false, false);
  }

#pragma unroll
  for (int r = 0; r < 8; ++r) {
    float* hrow = H + (size_t)(row0 + r + mD) * FDIM + n;
    hrow[0]  = acc0[r];
    hrow[16] = acc1[r];
    hrow[32] = acc2[r];
    hrow[48] = acc3[r];
  }
}

// ---------------------------------------------------------------------------
// Edge scatter: OUT[col[e]] += H[row[e]] * dinv[row]*dinv[col]
// 16 threads per edge, float4 gather, 4 f32 atomics each (L2-resident).
// ---------------------------------------------------------------------------
__global__ void scatter_edges(const float* __restrict__ H, const int* __restrict__ row,
                              const int* __restrict__ col, const float* __restrict__ dinv,
                              float* OUT, int e) {
  int t = blockIdx.x * blockDim.x + threadIdx.x;
  int eid = t >> 4;
  if (eid >= e) return;
  int f0 = (t & 15) * 4;
  int r = row[eid];
  int c = col[eid];
  float nrm = dinv[r] * dinv[c];
  const float4 hv = *(const float4*)(H + (size_t)r * FDIM + f0);
  float* o = OUT + (size_t)c * FDIM + f0;
  atomicAdd(o + 0, hv.x * nrm);
  atomicAdd(o + 1, hv.y * nrm);
  atomicAdd(o + 2, hv.z * nrm);
  atomicAdd(o + 3, hv.w * nrm);
}

// ---------------------------------------------------------------------------
// OUT[i] = relu(AGG[i] + H[i]*dinv[i]^2 + bias)   (self-loop fused; OUT may
// alias H or AGG — pure elementwise read-then-write.)
// ---------------------------------------------------------------------------
__global__ void bias_relu_self(const float* __restrict__ AGG, const float* __restrict__ H,
                               const float* __restrict__ dinv, const float* __restrict__ bias,
                               float* OUT, int n) {
  int t = blockIdx.x * blockDim.x + threadIdx.x;
  int i = t >> 4;
  if (i >= n) return;
  int f0 = (t & 15) * 4;
  float di = dinv[i];
  float s = di * di;
  const float4 a = *(const float4*)(AGG + (size_t)i * FDIM + f0);
  const float4 h = *(const float4*)(H + (size_t)i * FDIM + f0);
  const float4 b = *(const float4*)(bias + f0);
  float4 o;
  o.x = fmaxf(fmaf(h.x, s, a.x) + b.x, 0.0f);
  o.y = fmaxf(fmaf(h.y, s, a.y) + b.y, 0.0f);
  o.z = fmaxf(fmaf(h.z, s, a.z) + b.z, 0.0f);
  o.w = fmaxf(fmaf(h.w, s, a.w) + b.w, 0.0f);
  *(float4*)(OUT + (size_t)i * FDIM + f0) = o;
}

// ---------------------------------------------------------------------------
// Global add-pool: g[batch[i]] += H[i]
// ---------------------------------------------------------------------------
__global__ void pool_batch(const float* __restrict__ H, const int* __restrict__ batch,
                           float* g, int n) {
  int t = blockIdx.x * blockDim.x + threadIdx.x;
  int i = t >> 4;
  if (i >= n) return;
  int f0 = (t & 15) * 4;
  int b = batch[i];
  const float4 h = *(const float4*)(H + (size_t)i * FDIM + f0);
  float* o = g + (size_t)b * FDIM + f0;
  atomicAdd(o + 0, h.x);
  atomicAdd(o + 1, h.y);
  atomicAdd(o + 2, h.z);
  atomicAdd(o + 3, h.w);
}

// ---------------------------------------------------------------------------
// Readout: out[g] = dot(gpool[g], Wl) + bl
// ---------------------------------------------------------------------------
__global__ void readout(const float* __restrict__ gp, const float* __restrict__ Wl,
                        const float* __restrict__ bl, float* out, int G) {
  int i = blockIdx.x * blockDim.x + threadIdx.x;
  if (i >= G) return;
  float s = bl[0];
  const float* gv = gp + (size_t)i * FDIM;
#pragma unroll
  for (int k = 0; k < FDIM; ++k) s = fmaf(gv[k], Wl[k], s);
  out[i] = s;
}

// ---------------------------------------------------------------------------
extern "C" void kernel_launch(void* const* d_in, const int* in_sizes, int n_in,
                              void* d_out, int out_size, void* d_ws, size_t ws_size,
                              hipStream_t stream) {
  const float* x     = (const float*)d_in[0];
  const int*   eidx  = (const int*)d_in[1];
  const int*   batch = (const int*)d_in[2];
  const float* W1    = (const float*)d_in[3];
  const float* b1    = (const float*)d_in[4];
  const float* W2    = (const float*)d_in[5];
  const float* b2    = (const float*)d_in[6];
  const float* Wlin  = (const float*)d_in[7];
  const float* blin  = (const float*)d_in[8];
  float* out = (float*)d_out;

  const int N = in_sizes[0] / FDIM;   // 100000
  const int E = in_sizes[1] / 2;      // 1000000
  const int G = out_size;             // 1024
  const int* erow = eidx;
  const int* ecol = eidx + E;

  // Workspace partition (all f32): dinv[N] | bufA[N*64] | bufB[N*64] | gpool[G*64]
  float* dinv  = (float*)d_ws;
  float* bufA  = dinv + N;
  float* bufB  = bufA + (size_t)N * FDIM;
  float* gpool = bufB + (size_t)N * FDIM;

  const int TPB = 256;
  const size_t NF = (size_t)N * FDIM;
  const int gN    = (N + TPB - 1) / TPB;
  const int gE    = (E + TPB - 1) / TPB;
  const int gNF   = (int)((NF + TPB - 1) / TPB);
  const int gN16  = (N * 16 + TPB - 1) / TPB;   // node x 16-thread kernels
  const int gE16  = (E * 16 + TPB - 1) / TPB;   // edge x 16-thread kernels
  const int gGemm = (N + 127) / 128;            // 8 waves x 16 rows per block

  // --- degree & symmetric normalization ---
  init_ones<<<gN, TPB, 0, stream>>>(dinv, N);
  deg_count<<<gE, TPB, 0, stream>>>(ecol, dinv, E);
  finalize_dinv<<<gN, TPB, 0, stream>>>(dinv, N);

  // --- layer 1: h1 = relu(scatter(x@W1) + self + b1) -> bufA ---
  gemm64_wmma<<<gGemm, TPB, 0, stream>>>(x, W1, bufA, N);
  zero_f<<<gNF, TPB, 0, stream>>>(bufB, NF);
  scatter_edges<<<gE16, TPB, 0, stream>>>(bufA, erow, ecol, dinv, bufB, E);
  bias_relu_self<<<gN16, TPB, 0, stream>>>(bufB, bufA, dinv, b1, bufA, N);

  // --- layer 2: h2 = relu(scatter(h1@W2) + self + b2) -> bufB ---
  gemm64_wmma<<<gGemm, TPB, 0, stream>>>(bufA, W2, bufB, N);
  zero_f<<<gNF, TPB, 0, stream>>>(bufA, NF);
  scatter_edges<<<gE16, TPB, 0, stream>>>(bufB, erow, ecol, dinv, bufA, E);
  bias_relu_self<<<gN16, TPB, 0, stream>>>(bufA, bufB, dinv, b2, bufB, N);

  // --- pool + readout ---
  zero_f<<<(G * FDIM + TPB - 1) / TPB, TPB, 0, stream>>>(gpool, (size_t)G * FDIM);
  pool_batch<<<gN16, TPB, 0, stream>>>(bufB, batch, gpool, N);
  readout<<<(G + TPB - 1) / TPB, TPB, 0, stream>>>(gpool, Wlin, blin, out, G);
}